// DiscriminativeLoss_64785286693017
// MI455X (gfx1250) — compile-verified
//
#include <hip/hip_runtime.h>
#include <hip/hip_bf16.h>

typedef __attribute__((ext_vector_type(2))) float v2f;
typedef __attribute__((ext_vector_type(8))) float v8f;

#define F_DIM 16
#define K_DIM 24
#define DELTA_VAR 0.5f
#define DELTA_DIST 1.5f
#define EPS_ 1e-8f
#define GAMMA_ 0.001f
#define PIX_PER_WAVE 1024
#define WAVES_PER_BLOCK 8

// ---------------------------------------------------------------------------
// Kernel 1: per-batch GEMM  sums[f][c] = sum_l x[f,l] * t[c,l]  via fp32 WMMA
// (V_WMMA_F32_16X16X4_F32, two N-tiles for K=24), plus t_sum[c] and per-pixel
// label extraction via wave ballots (t is one-hot).
// One wave handles PIX_PER_WAVE pixels, 4 pixels (K=4) per WMMA step.
// Tile-1 lanes mn>=8 deliberately re-read rows 16..23 (same addresses
// coalesce); their garbage accumulator columns / ballot bits / ts1 partials
// are never consumed, so no masking is needed in the hot loop.
// ---------------------------------------------------------------------------
__global__ void k1_sums_wmma(const float* __restrict__ x,
                             const float* __restrict__ t,
                             float* __restrict__ sums,      // [B][16][24]
                             float* __restrict__ tsum,      // [B][24]
                             unsigned char* __restrict__ labels, // [B][L]
                             int L) {
    const int bb   = blockIdx.y;
    const int wave = threadIdx.x >> 5;
    const int lane = threadIdx.x & 31;
    const int g    = lane >> 4;      // half-wave group: K-offset 2*g
    const int mn   = lane & 15;      // A row (feature f) / B col (cluster c)
    const int kk   = 2 * g;

    const long l0 = (long)(blockIdx.x * WAVES_PER_BLOCK + wave) * PIX_PER_WAVE;
    if (l0 >= L) return;
    const long lend = min(l0 + (long)PIX_PER_WAVE, (long)L);

    const float* xr  = x + ((size_t)bb * F_DIM + mn) * L;           // A: feature row mn
    const float* tr0 = t + ((size_t)bb * K_DIM + mn) * L;           // B tile0: cluster mn
    const float* tr1 = t + ((size_t)bb * K_DIM + 16 + (mn & 7)) * L;// B tile1: cluster 16+(mn&7)
    unsigned char* labRow = labels + (size_t)bb * L;

    v8f acc0 = {};   // clusters 0..15
    v8f acc1 = {};   // clusters 16..23 in cols 0..7; cols 8..15 are unused dupes
    float ts0 = 0.f, ts1 = 0.f;

    for (long l = l0; l < lend; l += 16) {
        unsigned pk[4];
#pragma unroll
        for (int s = 0; s < 4; ++s) {
            const long lp = l + 4 * s;
            v2f a  = *(const v2f*)(xr  + lp + kk);
            v2f b0 = *(const v2f*)(tr0 + lp + kk);
            v2f b1 = *(const v2f*)(tr1 + lp + kk);

            acc0 = __builtin_amdgcn_wmma_f32_16x16x4_f32(false, a, false, b0,
                                                         (short)0, acc0, false, false);
            acc1 = __builtin_amdgcn_wmma_f32_16x16x4_f32(false, a, false, b1,
                                                         (short)0, acc1, false, false);

            ts0 += b0.x + b0.y;
            ts1 += b1.x + b1.y;   // lanes mn>=8 accumulate dupes; never stored

            // one-hot label via ballots (wave-uniform results -> SALU):
            // bit q of mA = t[q&15][lp + 2*(q>>4)]
            unsigned mA = __builtin_amdgcn_ballot_w32(b0.x > 0.5f);
            unsigned mB = __builtin_amdgcn_ballot_w32(b0.y > 0.5f);
            unsigned mC = __builtin_amdgcn_ballot_w32(b1.x > 0.5f);
            unsigned mD = __builtin_amdgcn_ballot_w32(b1.y > 0.5f);

            // duplicate bits 8..15 / 24..31 of mC,mD are masked right here
            unsigned w0 = (mA & 0xFFFFu) | ((mC & 0xFFu) << 16);          // pixel lp+0
            unsigned w1 = (mB & 0xFFFFu) | ((mD & 0xFFu) << 16);          // pixel lp+1
            unsigned w2 = (mA >> 16)     | (((mC >> 16) & 0xFFu) << 16);  // pixel lp+2
            unsigned w3 = (mB >> 16)     | (((mD >> 16) & 0xFFu) << 16);  // pixel lp+3

            pk[s] = (unsigned)__builtin_ctz(w0 | 0x01000000u)
                  | ((unsigned)__builtin_ctz(w1 | 0x01000000u) << 8)
                  | ((unsigned)__builtin_ctz(w2 | 0x01000000u) << 16)
                  | ((unsigned)__builtin_ctz(w3 | 0x01000000u) << 24);
        }
        if (lane == 0) {
            uint4 v;
            v.x = pk[0]; v.y = pk[1]; v.z = pk[2]; v.w = pk[3];
            *(uint4*)(labRow + l) = v;   // 16 labels, one b128 store
        }
    }

    // t_sum: combine the two half-wave partials for the same cluster column
    float tsA = ts0 + __shfl_xor(ts0, 16, 32);
    float tsB = ts1 + __shfl_xor(ts1, 16, 32);
    if (g == 0) {
        atomicAdd(&tsum[bb * K_DIM + mn], tsA);
        if (mn < 8) atomicAdd(&tsum[bb * K_DIM + 16 + mn], tsB);
    }

    // D tile: lane (g,mn), reg r -> f = r + 8*g, c = mn (tile0) / 16+mn (tile1)
#pragma unroll
    for (int r = 0; r < 8; ++r) {
        int f = r + 8 * g;
        atomicAdd(&sums[((size_t)bb * F_DIM + f) * K_DIM + mn], acc0[r]);
        if (mn < 8)
            atomicAdd(&sums[((size_t)bb * F_DIM + f) * K_DIM + 16 + mn], acc1[r]);
    }
}

// ---------------------------------------------------------------------------
// Kernel 2: per-batch means, m2, pairwise-distance hinge term, reg term.
// ---------------------------------------------------------------------------
__global__ void k2_means_dist(const float* __restrict__ sums,
                              const float* __restrict__ tsum,
                              const int* __restrict__ n_clusters,
                              float* __restrict__ means,   // [B][16][24]
                              float* __restrict__ m2,      // [B][24]
                              float* __restrict__ distreg) // [B]
{
    const int bb  = blockIdx.x;
    const int tid = threadIdx.x;
    const int nc  = n_clusters[bb];
    const float ncf = (float)nc;

    __shared__ float sm[F_DIM * K_DIM];
    __shared__ float sm2[K_DIM];
    __shared__ float red[256];

    for (int i = tid; i < F_DIM * K_DIM; i += blockDim.x) {
        int f = i / K_DIM, c = i % K_DIM;
        float v = 0.f;
        if (c < nc)
            v = sums[((size_t)bb * F_DIM + f) * K_DIM + c] /
                (tsum[bb * K_DIM + c] + EPS_);
        sm[f * K_DIM + c] = v;
        means[((size_t)bb * F_DIM + f) * K_DIM + c] = v;
    }
    __syncthreads();

    if (tid < K_DIM) {
        float s = 0.f;
#pragma unroll
        for (int f = 0; f < F_DIM; ++f) {
            float v = sm[f * K_DIM + tid];
            s += v * v;
        }
        sm2[tid] = s;
        m2[bb * K_DIM + tid] = s;
    }
    __syncthreads();

    // pairwise hinge
    float local = 0.f;
    for (int i = tid; i < K_DIM * K_DIM; i += blockDim.x) {
        int c = i / K_DIM, d = i % K_DIM;
        if (c != d && c < nc && d < nc) {
            float mm = 0.f;
#pragma unroll
            for (int f = 0; f < F_DIM; ++f)
                mm += sm[f * K_DIM + c] * sm[f * K_DIM + d];
            float pd2 = fmaxf(sm2[c] - 2.f * mm + sm2[d], 0.f);
            float pdist = (pd2 > 0.f) ? sqrtf(pd2) : 0.f;
            float h = fmaxf(2.f * DELTA_DIST - pdist, 0.f);
            local += h * h;
        }
    }
    red[tid] = local;
    __syncthreads();
    for (int s = blockDim.x >> 1; s > 0; s >>= 1) {
        if (tid < s) red[tid] += red[tid + s];
        __syncthreads();
    }
    float hingeSum = red[0];
    __syncthreads();

    // reg: sum over valid clusters of ||mu_c||
    float regl = 0.f;
    if (tid < K_DIM && tid < nc)
        regl = (sm2[tid] > 0.f) ? sqrtf(sm2[tid]) : 0.f;
    red[tid] = regl;
    __syncthreads();
    for (int s = blockDim.x >> 1; s > 0; s >>= 1) {
        if (tid < s) red[tid] += red[tid + s];
        __syncthreads();
    }

    if (tid == 0) {
        float dist_per = (nc > 1)
            ? hingeSum / (2.f * ncf * (ncf - 1.f) + EPS_) : 0.f;
        float reg_per = red[0] / ncf;
        distreg[bb] = dist_per + GAMMA_ * reg_per;   // BETA=1
    }
}

// ---------------------------------------------------------------------------
// Kernel 3: variance pass — per pixel, distance to its OWN cluster mean only
// (t is one-hot), using the 1-byte labels from kernel 1.
// ---------------------------------------------------------------------------
__global__ void k3_var(const float* __restrict__ x,
                       const unsigned char* __restrict__ labels,
                       const float* __restrict__ means,
                       const float* __restrict__ m2,
                       float* __restrict__ var_num,  // [B][24]
                       int L) {
    const int bb  = blockIdx.y;
    const int tid = threadIdx.x;

    __shared__ float smean[F_DIM * K_DIM];
    __shared__ float sm2[K_DIM];
    __shared__ float svar[K_DIM];

    for (int i = tid; i < F_DIM * K_DIM; i += blockDim.x)
        smean[i] = means[(size_t)bb * F_DIM * K_DIM + i];
    if (tid < K_DIM) { sm2[tid] = m2[bb * K_DIM + tid]; svar[tid] = 0.f; }
    __syncthreads();

    const size_t xbase = (size_t)bb * F_DIM * L;
    for (long l = (long)blockIdx.x * blockDim.x + tid; l < L;
         l += (long)gridDim.x * blockDim.x) {
        int c = labels[(size_t)bb * L + l];
        float x2 = 0.f, dot = 0.f;
#pragma unroll
        for (int f = 0; f < F_DIM; ++f) {
            float xv = x[xbase + (size_t)f * L + l];
            x2  += xv * xv;
            dot += xv * smean[f * K_DIM + c];
        }
        float d2 = fmaxf(x2 - 2.f * dot + sm2[c], 0.f);
        float dist = (d2 > 0.f) ? sqrtf(d2) : 0.f;
        float h = fmaxf(dist - DELTA_VAR, 0.f);
        atomicAdd(&svar[c], h * h);
    }
    __syncthreads();
    if (tid < K_DIM) atomicAdd(&var_num[bb * K_DIM + tid], svar[tid]);
}

// ---------------------------------------------------------------------------
// Kernel 4: fold everything into the output scalar.
// ---------------------------------------------------------------------------
__global__ void k4_final(const float* __restrict__ var_num,
                         const float* __restrict__ tsum,
                         const float* __restrict__ distreg,
                         const int* __restrict__ n_clusters,
                         float* __restrict__ out, int B) {
    if (threadIdx.x != 0 || blockIdx.x != 0) return;
    float total = 0.f;
    for (int bb = 0; bb < B; ++bb) {
        int nc = n_clusters[bb];
        float ncf = (float)nc;
        float vsum = 0.f;
        for (int c = 0; c < K_DIM; ++c)
            if (c < nc)
                vsum += var_num[bb * K_DIM + c] / (tsum[bb * K_DIM + c] + EPS_);
        float var_per = vsum / (ncf + EPS_);
        if (nc > 0) total += var_per;        // ALPHA=1
        total += distreg[bb];                 // BETA*dist + GAMMA*reg
    }
    out[0] = total / (float)B;
}

// ---------------------------------------------------------------------------
extern "C" void kernel_launch(void* const* d_in, const int* in_sizes, int n_in,
                              void* d_out, int out_size, void* d_ws, size_t ws_size,
                              hipStream_t stream) {
    const float* x = (const float*)d_in[0];
    const float* t = (const float*)d_in[1];
    const int* nclu = (const int*)d_in[2];

    const int B = in_sizes[2];
    const int L = in_sizes[0] / (B * F_DIM);   // 147456

    float* ws      = (float*)d_ws;
    float* sums    = ws;                       // B*16*24
    float* tsum    = sums    + (size_t)B * F_DIM * K_DIM;
    float* var_num = tsum    + (size_t)B * K_DIM;
    float* means   = var_num + (size_t)B * K_DIM;
    float* m2      = means   + (size_t)B * F_DIM * K_DIM;
    float* distreg = m2      + (size_t)B * K_DIM;
    unsigned char* labels = (unsigned char*)(distreg + B);

    // zero the atomic accumulation regions (sums, tsum, var_num)
    size_t zeroBytes = ((size_t)B * F_DIM * K_DIM + 2 * (size_t)B * K_DIM) * sizeof(float);
    hipMemsetAsync(d_ws, 0, zeroBytes, stream);

    const int pixPerBlock = PIX_PER_WAVE * WAVES_PER_BLOCK;
    dim3 g1((L + pixPerBlock - 1) / pixPerBlock, B);
    k1_sums_wmma<<<g1, 32 * WAVES_PER_BLOCK, 0, stream>>>(x, t, sums, tsum, labels, L);

    k2_means_dist<<<B, 256, 0, stream>>>(sums, tsum, nclu, means, m2, distreg);

    dim3 g3(72, B);
    k3_var<<<g3, 256, 0, stream>>>(x, labels, means, m2, var_num, L);

    k4_final<<<1, 32, 0, stream>>>(var_num, tsum, distreg, nclu, (float*)d_out, B);
}